// NMS_65111704207470
// MI455X (gfx1250) — compile-verified
//
#include <hip/hip_runtime.h>

// LDPC normalized-offset min-sum decoder for the wman_N0576_R34_z24 reference.
// B=256 frames, N=576, MR=144 checks, 15 edges/row, 3 iterations.
// Strategy: one workgroup per frame, everything LDS-resident, CDNA5 async
// global->LDS bulk copies for all read-only data, deterministic (atomic-free)
// column reductions via a CSC edge list built on-device.

#define B_   256
#define N_   576
#define MR_  144
#define Z_   24
#define IT_  3
#define DEG  15
#define NE   (MR_ * DEG)        // 2160 edges
#define NB_  (N_ / Z_)          // 24
#define AB   ((MR_ / Z_) * NB_ * IT_)  // 432 alpha/beta entries

// ---------------- CDNA5 async global->LDS helpers (gfx1250) ----------------
__device__ __forceinline__ void async_b128(void* lds, const void* g) {
  unsigned lo = (unsigned)(uintptr_t)lds;   // low 32 bits of flat ptr == LDS byte offset
  asm volatile("global_load_async_to_lds_b128 %0, %1, off"
               :: "v"(lo), "v"((unsigned long long)(uintptr_t)g) : "memory");
}
__device__ __forceinline__ void async_b32(void* lds, const void* g) {
  unsigned lo = (unsigned)(uintptr_t)lds;
  asm volatile("global_load_async_to_lds_b32 %0, %1, off"
               :: "v"(lo), "v"((unsigned long long)(uintptr_t)g) : "memory");
}
__device__ __forceinline__ void async_wait0() {
  asm volatile("s_wait_asynccnt 0x0" ::: "memory");
}

// ------------------------------- decoder ------------------------------------
// (placed first in the TU so the disasm snippet shows the async-to-LDS path)
__global__ __launch_bounds__(256) void ldpc_kernel(
    const float* __restrict__ r, const float* __restrict__ alpha,
    const float* __restrict__ beta, const int* __restrict__ cols,
    const int* __restrict__ colptr, const int* __restrict__ coledge,
    float* __restrict__ out) {
  __shared__ float r_s[N_];
  __shared__ float sum_s[N_];
  __shared__ float M_s[NE];
  __shared__ float E_s[NE];
  __shared__ int   c_s[NE];
  __shared__ int   ce_s[NE];
  __shared__ int   cp_s[N_ + 1];
  __shared__ float a_s[AB];
  __shared__ float b_s[AB];

  const int b = blockIdx.x;
  const int t = threadIdx.x;
  const int NT = 256;

  // ---- async bulk copy all read-only data into LDS (b128 = 16B/lane) ----
  for (int i = t; i < N_ / 4; i += NT)
    async_b128(&r_s[i * 4], r + (size_t)b * N_ + i * 4);
  for (int i = t; i < NE / 4; i += NT)
    async_b128(&c_s[i * 4], cols + i * 4);
  for (int i = t; i < NE / 4; i += NT)
    async_b128(&ce_s[i * 4], coledge + i * 4);
  for (int i = t; i < (N_ + 1) / 4; i += NT)   // 144 x b128 covers [0,576)
    async_b128(&cp_s[i * 4], colptr + i * 4);
  if (t == 0) async_b32(&cp_s[N_], colptr + N_);
  for (int i = t; i < AB / 4; i += NT)
    async_b128(&a_s[i * 4], alpha + i * 4);
  for (int i = t; i < AB / 4; i += NT)
    async_b128(&b_s[i * 4], beta + i * 4);
  async_wait0();
  __syncthreads();

  // ---- init v->c messages at edges: M = r[col] ----
  for (int e = t; e < NE; e += NT) M_s[e] = r_s[c_s[e]];
  __syncthreads();

  for (int it = 0; it < IT_; ++it) {
    // ---- check-node update: min-2 magnitude + sign product per row ----
    if (t < MR_) {
      const int m = t;
      const int base = m * DEG;                // stride 15: coprime to 64 banks
      const int mb = m / Z_;
      float sg[DEG];
      int   cj[DEG];
      float m1 = __builtin_inff(), m2 = __builtin_inff();
      int   arg = 0;
      float prod = 1.0f;
#pragma unroll
      for (int j = 0; j < DEG; ++j) {
        float v = M_s[base + j];
        cj[j] = c_s[base + j];
        float a = fabsf(v);
        float s = (v > 0.0f) ? 1.0f : ((v < 0.0f) ? -1.0f : 0.0f);  // sign(0)=0
        sg[j] = s;
        prod *= s;                              // any zero kills the whole row
        if (a < m1) { m2 = m1; m1 = a; arg = j; }
        else if (a < m2) { m2 = a; }
      }
#pragma unroll
      for (int j = 0; j < DEG; ++j) {
        const int ab = (mb * NB_ + cj[j] / Z_) * IT_ + it;
        float ea  = (j == arg) ? m2 : m1;       // exclude-self min
        float mag = ea - b_s[ab];
        mag = (mag > 0.0f) ? mag : 0.0f;        // relu(E_abs - beta)
        E_s[base + j] = a_s[ab] * (prod * sg[j]) * mag;
      }
    }
    __syncthreads();

    // ---- variable-node: per-column sum of E in fixed order (no atomics) ----
    for (int n = t; n < N_; n += NT) {
      float s = 0.0f;
      const int e0 = cp_s[n], e1 = cp_s[n + 1];
      for (int k = e0; k < e1; ++k) s += E_s[ce_s[k]];
      sum_s[n] = s;
    }
    __syncthreads();

    // ---- v->c messages: sum-minus-self (skip on last iteration) ----
    if (it != IT_ - 1) {
      for (int e = t; e < NE; e += NT) {
        const int n = c_s[e];
        M_s[e] = r_s[n] + sum_s[n] - E_s[e];
      }
      __syncthreads();
    }
  }

  // ---- posterior LLRs: vectorized b128 stores (16B-aligned frame base) ----
  float4* out4 = (float4*)(out + (size_t)b * N_);
  for (int i = t; i < N_ / 4; i += NT) {
    float4 v;
    v.x = r_s[i * 4 + 0] + sum_s[i * 4 + 0];
    v.y = r_s[i * 4 + 1] + sum_s[i * 4 + 1];
    v.z = r_s[i * 4 + 2] + sum_s[i * 4 + 2];
    v.w = r_s[i * 4 + 3] + sum_s[i * 4 + 3];
    out4[i] = v;
  }
}

// -------- preprocessing: row adjacency (15 column indices per check row) ----
__global__ __launch_bounds__(64) void build_rows(const int* __restrict__ H,
                                                 int* __restrict__ cols) {
  int m = blockIdx.x * blockDim.x + threadIdx.x;
  if (m >= MR_) return;
  int k = 0;
  for (int n = 0; n < N_; ++n) {
    if (H[m * N_ + n] != 0 && k < DEG) cols[m * DEG + k++] = n;
  }
}

// -------- preprocessing: CSC edge list for deterministic column sums --------
__global__ __launch_bounds__(N_) void build_csc(const int* __restrict__ cols,
                                                int* __restrict__ colptr,
                                                int* __restrict__ coledge) {
  __shared__ int ptr_s[N_ + 1];
  const int n = threadIdx.x;                  // blockDim.x == N_
  int cnt = 0;
  for (int e = 0; e < NE; ++e) cnt += (cols[e] == n) ? 1 : 0;
  ptr_s[n] = cnt;
  __syncthreads();
  if (n == 0) {
    int acc = 0;
    for (int i = 0; i < N_; ++i) { int c = ptr_s[i]; ptr_s[i] = acc; acc += c; }
    ptr_s[N_] = acc;
  }
  __syncthreads();
  int w = ptr_s[n];
  colptr[n] = w;
  if (n == 0) colptr[N_] = ptr_s[N_];
  for (int e = 0; e < NE; ++e)
    if (cols[e] == n) coledge[w++] = e;       // fixed edge order -> deterministic sums
}

extern "C" void kernel_launch(void* const* d_in, const int* in_sizes, int n_in,
                              void* d_out, int out_size, void* d_ws, size_t ws_size,
                              hipStream_t stream) {
  (void)in_sizes; (void)n_in; (void)out_size; (void)ws_size;
  const float* r     = (const float*)d_in[0];
  const float* alpha = (const float*)d_in[1];
  const float* beta  = (const float*)d_in[2];
  const int*   H     = (const int*)d_in[3];
  float* out = (float*)d_out;

  // workspace layout (all 16B aligned): cols | coledge | colptr
  char* ws = (char*)d_ws;
  int* cols    = (int*)(ws);                  // NE ints   (8640 B)
  int* coledge = (int*)(ws + 8640);           // NE ints   (8640 B)
  int* colptr  = (int*)(ws + 17280);          // N_+1 ints (2308 B)

  build_rows<<<(MR_ + 63) / 64, 64, 0, stream>>>(H, cols);
  build_csc<<<1, N_, 0, stream>>>(cols, colptr, coledge);
  ldpc_kernel<<<B_, 256, 0, stream>>>(r, alpha, beta, cols, colptr, coledge, out);
}